// affine_transformation_37082747634496
// MI455X (gfx1250) — compile-verified
//
#include <hip/hip_runtime.h>
#include <math.h>

// Affine grid + bilinear grid_sample, B=64 C=3 H=W=512, fp32.
// Bandwidth-bound gather: no WMMA formulation reduces bytes moved.
// gfx1250-specific paths used: global_prefetch_b8 (next-row gather prefetch),
// non-temporal global_store_b128 (write-once output stream, protect L2),
// hardware v_sin_f32/v_cos_f32 for the per-thread affine setup.

typedef float v4f __attribute__((ext_vector_type(4)));

#define HDIM 512
#define WDIM 512
#define CH   3
#define RPB  4     // rows per block
#define TPB  128   // threads: each covers 4 consecutive w -> 128*4 = 512 = W

__global__ __launch_bounds__(TPB) void affine_bilinear_kernel(
    const float* __restrict__ x, const float* __restrict__ mp,
    float* __restrict__ out)
{
  const int b  = blockIdx.y;                 // wave-uniform -> scalar param loads
  const int h0 = blockIdx.x * RPB;
  const int w0 = (int)threadIdx.x << 2;

  const float theta = mp[b * 4 + 0];
  const float scl   = mp[b * 4 + 1];
  const float txp   = mp[b * 4 + 2];
  const float typ   = mp[b * 4 + 3];
  // theta ~ N(0,1): hardware trig (v_sin_f32/v_cos_f32) is plenty accurate
  // and avoids dragging in the scalarized Payne-Hanek libm path.
  const float ct = scl * __cosf(theta);      // A00 = A11
  const float st = scl * __sinf(theta);      // A10 = -A01

  const size_t HW = (size_t)HDIM * WDIM;
  const float* __restrict__ p0 = x + ((size_t)b * CH) * HW;
  const float* __restrict__ p1 = p0 + HW;
  const float* __restrict__ p2 = p1 + HW;
  float* __restrict__ o0 = out + ((size_t)b * CH) * HW;
  float* __restrict__ o1 = o0 + HW;
  float* __restrict__ o2 = o1 + HW;

  // xs = (2w+1)/W - 1 for the 4 pixels this thread owns
  float xs[4];
#pragma unroll
  for (int j = 0; j < 4; ++j)
    xs[j] = fmaf(2.0f / WDIM, (float)(w0 + j), 1.0f / WDIM - 1.0f);

  for (int r = 0; r < RPB; ++r) {
    const int   h  = h0 + r;
    const float ys = fmaf(2.0f / HDIM, (float)h, 1.0f / HDIM - 1.0f);
    const float gxr = fmaf(-st, ys, txp);    // row-constant part of gx
    const float gyr = fmaf( ct, ys, typ);    // row-constant part of gy

    v4f a0, a1, a2;
#pragma unroll
    for (int j = 0; j < 4; ++j) {
      const float gx = fmaf(ct, xs[j], gxr);
      const float gy = fmaf(st, xs[j], gyr);
      // ix = ((gx+1)*W - 1)*0.5 = gx*W/2 + (W-1)/2
      const float ix = fmaf(gx, 0.5f * WDIM, 0.5f * WDIM - 0.5f);
      const float iy = fmaf(gy, 0.5f * HDIM, 0.5f * HDIM - 0.5f);

      const float x0f = floorf(ix);
      const float y0f = floorf(iy);
      const float fx  = ix - x0f;
      const float fy  = iy - y0f;
      const int xi0 = (int)x0f, yi0 = (int)y0f;
      const int xi1 = xi0 + 1,  yi1 = yi0 + 1;

      // validity masks on UNclamped indices (matches reference), then clamp
      const float vx0 = ((unsigned)xi0 < (unsigned)WDIM) ? 1.0f : 0.0f;
      const float vx1 = ((unsigned)xi1 < (unsigned)WDIM) ? 1.0f : 0.0f;
      const float vy0 = ((unsigned)yi0 < (unsigned)HDIM) ? 1.0f : 0.0f;
      const float vy1 = ((unsigned)yi1 < (unsigned)HDIM) ? 1.0f : 0.0f;
      const int xc0 = min(max(xi0, 0), WDIM - 1);
      const int xc1 = min(max(xi1, 0), WDIM - 1);
      const int yc0 = min(max(yi0, 0), HDIM - 1);
      const int yc1 = min(max(yi1, 0), HDIM - 1);

      const float w00 = (1.0f - fx) * (1.0f - fy) * (vy0 * vx0);
      const float w01 = fx * (1.0f - fy) * (vy0 * vx1);
      const float w10 = (1.0f - fx) * fy * (vy1 * vx0);
      const float w11 = fx * fy * (vy1 * vx1);

      const int i00 = yc0 * WDIM + xc0;
      const int i01 = yc0 * WDIM + xc1;
      const int i10 = yc1 * WDIM + xc0;
      const int i11 = yc1 * WDIM + xc1;

      a0[j] = fmaf(w00, p0[i00], fmaf(w01, p0[i01], fmaf(w10, p0[i10], w11 * p0[i11])));
      a1[j] = fmaf(w00, p1[i00], fmaf(w01, p1[i01], fmaf(w10, p1[i10], w11 * p1[i11])));
      a2[j] = fmaf(w00, p2[i00], fmaf(w01, p2[i01], fmaf(w10, p2[i10], w11 * p2[i11])));

      // Next row's source moves by exactly (dix, diy) = (-st, +ct) pixels
      // (W==H). Prefetch its gather neighborhood -> global_prefetch_b8.
      if (j == 0 && r + 1 < RPB) {
        const int xpn = min(max((int)floorf(ix - st), 0), WDIM - 1);
        const int ypn = min(max((int)floorf(iy + ct), 0), HDIM - 1);
        const long pn = (long)ypn * WDIM + xpn;
        __builtin_prefetch(p0 + pn, 0, 1);
        __builtin_prefetch(p1 + pn, 0, 1);
        __builtin_prefetch(p2 + pn, 0, 1);
      }
    }

    // Write-once output: 128-bit coalesced non-temporal stores keep the
    // gather stream's source data resident in L2.
    const size_t off = (size_t)h * WDIM + (size_t)w0;
    __builtin_nontemporal_store(a0, (v4f*)(o0 + off));
    __builtin_nontemporal_store(a1, (v4f*)(o1 + off));
    __builtin_nontemporal_store(a2, (v4f*)(o2 + off));
  }
}

extern "C" void kernel_launch(void* const* d_in, const int* in_sizes, int n_in,
                              void* d_out, int out_size, void* d_ws, size_t ws_size,
                              hipStream_t stream) {
  const float* x  = (const float*)d_in[0];   // (64, 3, 512, 512) f32
  const float* mp = (const float*)d_in[1];   // (64, 4) f32
  float* out = (float*)d_out;                // (64, 3, 512, 512) f32

  const int B = in_sizes[1] / 4;             // 64
  dim3 grid(HDIM / RPB, B);                  // (128, 64) blocks
  affine_bilinear_kernel<<<grid, TPB, 0, stream>>>(x, mp, out);
}